// GATLayer_46067819217388
// MI455X (gfx1250) — compile-verified
//
#include <hip/hip_runtime.h>
#include <math.h>

#define GN 4096
#define GIN_F 512
#define GOUT_F 256
#define GHEADS 4
#define GHEAD_DIM 64

typedef __attribute__((ext_vector_type(2))) float v2f;
typedef __attribute__((ext_vector_type(8))) float v8f;

// ---------------------------------------------------------------------------
// Kernel 1: H = X @ W^T   (M=4096, N=256, K=512) in fp32 via v_wmma_f32_16x16x4_f32
// Block = 512 threads = 16 waves; each wave owns one 16x16 output tile
// (16 i-rows shared by the block, 16 features per wave). K advances by 4.
// A layout (ISA 7.12.2, 32-bit A 16x4): lanes 0-15 hold K={0,1}, lanes 16-31 K={2,3}.
// B layout mirrors: lanes 0-15 hold rows K={0,1} at column N=lane, lanes 16-31 K={2,3}.
// ---------------------------------------------------------------------------
__global__ __launch_bounds__(512) void gat_gemm_h(const float* __restrict__ X,
                                                  const float* __restrict__ W,
                                                  float* __restrict__ H) {
    const int wave = threadIdx.x >> 5;
    const int lane = threadIdx.x & 31;
    const int i0   = blockIdx.x * 16;
    const int f0   = wave * 16;
    const int col  = lane & 15;          // M for A, N for B/C
    const int koff = (lane >> 4) * 2;    // K sub-offset per half-wave

    v8f acc = {};
    for (int k = 0; k < GIN_F; k += 4) {
        v2f a, b;
        const float* xr = X + (size_t)(i0 + col) * GIN_F + k + koff;
        a.x = xr[0];
        a.y = xr[1];
        // B[k][n] = W[n][k]
        const float* wr = W + (size_t)(f0 + col) * GIN_F + k + koff;
        b.x = wr[0];
        b.y = wr[1];
        acc = __builtin_amdgcn_wmma_f32_16x16x4_f32(false, a, false, b,
                                                    (short)0, acc, false, false);
    }
    const int rbase = (lane >> 4) * 8;   // C/D layout: lanes 16-31 hold M=8..15
    for (int r = 0; r < 8; ++r)
        H[(size_t)(i0 + rbase + r) * GOUT_F + f0 + col] = acc[r];
}

// ---------------------------------------------------------------------------
// Kernel 2: e_src[n,h] = <h[n,h,:], a[h,:D]>, e_dst[n,h] = <h[n,h,:], a[h,D:]>
// ---------------------------------------------------------------------------
__global__ void gat_edge_terms(const float* __restrict__ H,
                               const float* __restrict__ A,
                               float* __restrict__ ES,
                               float* __restrict__ ED) {
    const int t = blockIdx.x * blockDim.x + threadIdx.x;
    if (t >= GN * GHEADS) return;
    const int n  = t >> 2;
    const int hd = t & 3;
    const float* hv = H + (size_t)n * GOUT_F + hd * GHEAD_DIM;
    const float* as = A + hd * 2 * GHEAD_DIM;
    const float* ad = as + GHEAD_DIM;
    float s = 0.f, d = 0.f;
    for (int dd = 0; dd < GHEAD_DIM; ++dd) {
        const float v = hv[dd];
        s += v * as[dd];
        d += v * ad[dd];
    }
    ES[t] = s;
    ED[t] = d;
}

// ---------------------------------------------------------------------------
// Kernel 3: per-row softmax stats. One block per row i.
// m[i,h]  = max_j masked lrelu(s_ih + d_jh)
// il[i,h] = 0.25 / sum_j exp(.. - m)   (0 if the row is fully masked -> nan_to_num)
// ---------------------------------------------------------------------------
__global__ __launch_bounds__(256) void gat_row_stats(const float* __restrict__ ADJ,
                                                     const float* __restrict__ ES,
                                                     const float* __restrict__ ED,
                                                     float* __restrict__ M,
                                                     float* __restrict__ IL) {
    __shared__ float red[GHEADS][256];
    const int i   = blockIdx.x;
    const int tid = threadIdx.x;
    const float* arow = ADJ + (size_t)i * GN;

    float s[GHEADS];
    for (int h = 0; h < GHEADS; ++h) s[h] = ES[i * GHEADS + h];

    // pass 1: max
    float mx[GHEADS];
    for (int h = 0; h < GHEADS; ++h) mx[h] = -INFINITY;
    for (int j = tid; j < GN; j += 256) {
        if (arow[j] > 0.1f) {
            for (int h = 0; h < GHEADS; ++h) {
                float v = s[h] + ED[j * GHEADS + h];
                v = (v >= 0.f) ? v : 0.2f * v;
                mx[h] = fmaxf(mx[h], v);
            }
        }
    }
    for (int h = 0; h < GHEADS; ++h) red[h][tid] = mx[h];
    __syncthreads();
    for (int off = 128; off > 0; off >>= 1) {
        if (tid < off)
            for (int h = 0; h < GHEADS; ++h)
                red[h][tid] = fmaxf(red[h][tid], red[h][tid + off]);
        __syncthreads();
    }
    float m[GHEADS];
    for (int h = 0; h < GHEADS; ++h) m[h] = red[h][0];
    __syncthreads();

    // pass 2: sum of exp (adj row is L0/L2 resident after pass 1)
    float sm[GHEADS] = {0.f, 0.f, 0.f, 0.f};
    for (int j = tid; j < GN; j += 256) {
        if (arow[j] > 0.1f) {
            for (int h = 0; h < GHEADS; ++h) {
                float v = s[h] + ED[j * GHEADS + h];
                v = (v >= 0.f) ? v : 0.2f * v;
                sm[h] += __expf(v - m[h]);
            }
        }
    }
    for (int h = 0; h < GHEADS; ++h) red[h][tid] = sm[h];
    __syncthreads();
    for (int off = 128; off > 0; off >>= 1) {
        if (tid < off)
            for (int h = 0; h < GHEADS; ++h)
                red[h][tid] += red[h][tid + off];
        __syncthreads();
    }
    if (tid < GHEADS) {
        const float l = red[tid][0];
        M[i * GHEADS + tid]  = m[tid];
        IL[i * GHEADS + tid] = (l > 0.f) ? (0.25f / l) : 0.f;  // folds head-mean
    }
}

// ---------------------------------------------------------------------------
// Kernel 4: OUT = P @ H + bias, 32-row i-tiles; the 32x64 P tile is built on
// the fly in LDS. Block = 512 threads = 16 waves; wave w owns feature columns
// [16w, 16w+16) for BOTH 16-row halves of the tile (two accumulators sharing
// every B fragment -> halves L2 traffic for H). j streamed in chunks of 64,
// with global_prefetch of the next adj chunk.
// ---------------------------------------------------------------------------
#define MTILE 32

__global__ __launch_bounds__(512) void gat_attn_out(const float* __restrict__ ADJ,
                                                    const float* __restrict__ H,
                                                    const float* __restrict__ ES,
                                                    const float* __restrict__ ED,
                                                    const float* __restrict__ M,
                                                    const float* __restrict__ IL,
                                                    const float* __restrict__ BIAS,
                                                    float* __restrict__ OUT) {
    __shared__ float sP[MTILE * 64];        // attention tile (row-major, ld=64)
    __shared__ float sS[MTILE * GHEADS];
    __shared__ float sM[MTILE * GHEADS];
    __shared__ float sIL[MTILE * GHEADS];
    __shared__ float sD[64 * GHEADS];       // e_dst for current j-chunk

    const int tid  = threadIdx.x;
    const int wave = tid >> 5;
    const int lane = tid & 31;
    const int i0   = blockIdx.x * MTILE;
    const int f0   = wave * 16;
    const int col  = lane & 15;
    const int koff = (lane >> 4) * 2;

    if (tid < MTILE * GHEADS) {
        sS[tid]  = ES[i0 * GHEADS + tid];
        sM[tid]  = M[i0 * GHEADS + tid];
        sIL[tid] = IL[i0 * GHEADS + tid];
    }

    v8f acc0 = {};   // rows i0 .. i0+15
    v8f acc1 = {};   // rows i0+16 .. i0+31
    for (int jc = 0; jc < GN; jc += 64) {
        __syncthreads();                    // sP/sD free to overwrite
        if (tid < 64 * GHEADS) sD[tid] = ED[jc * GHEADS + tid];

        // prefetch next adj chunk (32 rows x 256B) while we crunch this one
        const int jcn = jc + 64;
        if (jcn < GN && tid < MTILE * 2) {
            const int prow = tid >> 1;
            const int poff = (tid & 1) * 32;
            __builtin_prefetch(&ADJ[(size_t)(i0 + prow) * GN + jcn + poff], 0, 1);
        }
        __syncthreads();

        // build P[32][64]: 2048 entries, 4 per thread, coalesced over j
        for (int e = tid; e < MTILE * 64; e += 512) {
            const int row = e >> 6;
            const int j   = e & 63;
            const float adjv = ADJ[(size_t)(i0 + row) * GN + jc + j];
            float p = 0.f;
            if (adjv > 0.1f) {
                for (int h = 0; h < GHEADS; ++h) {
                    float v = sS[row * GHEADS + h] + sD[j * GHEADS + h];
                    v = (v >= 0.f) ? v : 0.2f * v;
                    p += __expf(v - sM[row * GHEADS + h]) * sIL[row * GHEADS + h];
                }
            }
            sP[row * 64 + j] = p;
        }
        __syncthreads();

        // 32 fp32 WMMAs over this chunk: A fragments from LDS, one shared B
        // fragment (H rows, L2 resident) feeds both row-halves.
        for (int ks = 0; ks < 64; ks += 4) {
            v2f a0, a1, b;
            a0.x = sP[col * 64 + ks + koff];
            a0.y = sP[col * 64 + ks + koff + 1];
            a1.x = sP[(col + 16) * 64 + ks + koff];
            a1.y = sP[(col + 16) * 64 + ks + koff + 1];
            const int jrow = jc + ks + koff;
            b.x = H[(size_t)jrow * GOUT_F + f0 + col];
            b.y = H[(size_t)(jrow + 1) * GOUT_F + f0 + col];
            acc0 = __builtin_amdgcn_wmma_f32_16x16x4_f32(false, a0, false, b,
                                                         (short)0, acc0, false, false);
            acc1 = __builtin_amdgcn_wmma_f32_16x16x4_f32(false, a1, false, b,
                                                         (short)0, acc1, false, false);
        }
    }

    const int rbase = (lane >> 4) * 8;
    const float bv = BIAS[f0 + col];
    for (int r = 0; r < 8; ++r) {
        OUT[(size_t)(i0 + rbase + r) * GOUT_F + f0 + col]      = acc0[r] + bv;
        OUT[(size_t)(i0 + 16 + rbase + r) * GOUT_F + f0 + col] = acc1[r] + bv;
    }
}

// ---------------------------------------------------------------------------
extern "C" void kernel_launch(void* const* d_in, const int* in_sizes, int n_in,
                              void* d_out, int out_size, void* d_ws, size_t ws_size,
                              hipStream_t stream) {
    const float* features = (const float*)d_in[0];   // [4096, 512]
    const float* adj      = (const float*)d_in[1];   // [4096, 4096]
    const float* W        = (const float*)d_in[2];   // [256, 512]
    const float* attw     = (const float*)d_in[3];   // [4, 128]
    const float* bias     = (const float*)d_in[4];   // [256]
    float* out = (float*)d_out;                      // [4096, 256]

    float* ws  = (float*)d_ws;
    float* H   = ws;                                 // 4096*256 floats
    float* ES  = H  + (size_t)GN * GOUT_F;           // 4096*4
    float* ED  = ES + GN * GHEADS;                   // 4096*4
    float* Mb  = ED + GN * GHEADS;                   // 4096*4
    float* ILb = Mb + GN * GHEADS;                   // 4096*4

    gat_gemm_h    <<<GN / 16,              512, 0, stream>>>(features, W, H);
    gat_edge_terms<<<(GN * GHEADS) / 256,  256, 0, stream>>>(H, attw, ES, ED);
    gat_row_stats <<<GN,                   256, 0, stream>>>(adj, ES, ED, Mb, ILb);
    gat_attn_out  <<<GN / MTILE,           512, 0, stream>>>(adj, H, ES, ED, Mb, ILb, bias, out);
}